// BSRNNMaskGenerator_75823352644256
// MI455X (gfx1250) — compile-verified
//
#include <hip/hip_runtime.h>
#include <hip/hip_bf16.h>

#define NB     31
#define BB     8
#define CC     128
#define TT     1000
#define MAXBW  16
#define MAXOC  128
#define ENCDIM 257
#define EPSF   1e-8f
#define TILE_T 64
#define WPADN  130   // W tile row stride [o][c]; even -> 8B-aligned K pairs
#define XPADN  130   // X tile row stride [t][c]; even -> 8B-aligned K pairs (b64 loads)
#define YPADN  132   // Y tile row stride [t][o]; mult of 4 -> 16B-aligned b128 stores

typedef float v2f __attribute__((ext_vector_type(2)));
typedef float v4f __attribute__((ext_vector_type(4)));
typedef float v8f __attribute__((ext_vector_type(8)));

__device__ __constant__ int d_boff[NB] = {
    0,3,6,9,12,15,18,21,24,27,
    30,38,46,54,62,70,78,86,94,102,110,118,
    126,142,158,174,190,206,222,238,
    254};

// ---------------------------------------------------------------------------
// Kernel 1: per-(band,batch) mean / rstd over (C,T)
// ---------------------------------------------------------------------------
__global__ __launch_bounds__(256) void bsrnn_stats_kernel(
    const float* __restrict__ sep, float* __restrict__ stats)
{
    const int blk  = blockIdx.x;        // band*BB + b
    const int band = blk / BB;
    const int b    = blk % BB;
    const int tid  = threadIdx.x;

    float s = 0.f, q = 0.f;
    for (int j = tid; j < CC * TT; j += 256) {
        const int c = j / TT;
        const int t = j - c * TT;
        const float x = sep[((size_t)(b * CC + c) * TT + t) * NB + band];
        s += x;
        q += x * x;
    }
    __shared__ float sh_s[256];
    __shared__ float sh_q[256];
    sh_s[tid] = s; sh_q[tid] = q;
    __syncthreads();
    for (int off = 128; off > 0; off >>= 1) {
        if (tid < off) {
            sh_s[tid] += sh_s[tid + off];
            sh_q[tid] += sh_q[tid + off];
        }
        __syncthreads();
    }
    if (tid == 0) {
        const float inv  = 1.0f / (float)(CC * TT);
        const float mean = sh_s[0] * inv;
        const float var  = sh_q[0] * inv - mean * mean;
        stats[2 * blk]     = mean;
        stats[2 * blk + 1] = rsqrtf(var + EPSF);
    }
}

// ---------------------------------------------------------------------------
// Kernel 2 (templated on band width: WMMA accumulator count is compile-time).
// W tile [o][c] stride 130, X tile [t][c] stride 130 -> both A and B
// fragments are single 8B-aligned ds_load_b64. Y tile [t][o] stride 132 ->
// epilogue spills via 16B-aligned ds_store_b128 pairs.
// ---------------------------------------------------------------------------
template <int BW>
__global__ __launch_bounds__(256) void bsrnn_gemm_mask_kernel(
    const float* __restrict__ sep,
    const float* __restrict__ ctx_r,
    const float* __restrict__ ctx_i,
    const float* __restrict__ gamma,
    const float* __restrict__ beta,
    const float* __restrict__ convw,
    const float* __restrict__ convb,
    const float* __restrict__ stats,
    float* __restrict__ out,
    int band_base)
{
    constexpr int OC     = BW * 8;              // 24 / 64 / 128
    constexpr int MTILES = (OC + 15) >> 4;      // 2  / 4  / 8
    constexpr int NACC   = (MTILES * 4) / 8;    // 1  / 2  / 4
    constexpr int HALFOC = OC / 2;              // 12 / 32 / 64

    extern __shared__ float smem[];
    float* Ws = smem;                       // [MTILES*16][WPADN]
    float* Xs = smem + MAXOC * WPADN;       // [TILE_T][XPADN]  (reused as Ys [TILE_T][YPADN])

    const int band = band_base + blockIdx.z;
    const int b    = blockIdx.y;
    const int t0   = blockIdx.x * TILE_T;
    const int tlen = min(TILE_T, TT - t0);

    const int tid  = threadIdx.x;
    const int lane = tid & 31;
    const int w    = tid >> 5;              // wave id 0..7
    const int lm   = lane & 15;
    const int lh   = lane >> 4;             // 0: K0/K1 half, 1: K2/K3 half

    const float mean = stats[2 * (band * BB + b)];
    const float rstd = stats[2 * (band * BB + b) + 1];

    // ---- stage W tile [o][c] (rows [0, MTILES*16)) ----
    for (int idx = tid; idx < MTILES * 16 * CC; idx += 256) {
        const int o = idx >> 7;             // CC == 128
        const int c = idx & 127;
        Ws[o * WPADN + c] = convw[((size_t)band * MAXOC + o) * CC + c];
    }
    // ---- stage normalized X tile transposed: [tt][c] ----
    for (int idx = tid; idx < TILE_T * CC; idx += 256) {
        const int tt = idx >> 7;            // CC == 128
        const int c  = idx & 127;
        float v = 0.f;
        if (tt < tlen) {
            const float x = sep[((size_t)(b * CC + c) * TT + (t0 + tt)) * NB + band];
            v = (x - mean) * rstd * gamma[band * CC + c] + beta[band * CC + c];
        }
        Xs[tt * XPADN + c] = v;
    }
    __syncthreads();

    // ---- WMMA K-loop: subtile s = w + 8*j ; mtile = s>>2, nsub = s&3 ----
    v8f acc[NACC];
    #pragma unroll
    for (int j = 0; j < NACC; ++j)
        acc[j] = (v8f){0.f,0.f,0.f,0.f,0.f,0.f,0.f,0.f};

    const int kkh = lh << 1;                // 0 or 2: this half-wave's K pair base
    for (int ks = 0; ks < CC; ks += 4) {
        #pragma unroll
        for (int j = 0; j < NACC; ++j) {
            const int s     = w + (j << 3);
            const int mtile = s >> 2;
            const int nsub  = s & 3;
            // A fragment (16x4 fp32, 2 VGPRs): one aligned b64 load
            const int mrow = (mtile << 4) + lm;
            const v2f a = *(const v2f*)&Ws[mrow * WPADN + ks + kkh];
            // B fragment (4x16 fp32, 2 VGPRs): one aligned b64 load
            const int tcol = (nsub << 4) + lm;
            const v2f bf = *(const v2f*)&Xs[tcol * XPADN + ks + kkh];
            acc[j] = __builtin_amdgcn_wmma_f32_16x16x4_f32(
                /*neg_a=*/false, a, /*neg_b=*/false, bf,
                /*c_mod=*/(short)0, acc[j],
                /*reuse_a=*/false, /*reuse_b=*/false);
        }
    }
    __syncthreads();                         // all waves done with Xs

    // ---- spill y + bias into LDS as Ys[t][o] via b128 stores ----
    float* Ys = Xs;
    #pragma unroll
    for (int j = 0; j < NACC; ++j) {
        const int s     = w + (j << 3);
        const int mtile = s >> 2;
        const int nsub  = s & 3;
        const int col   = (nsub << 4) + lm;          // t within tile
        const int base0 = (mtile << 4) + (lh << 3);  // first of 8 consecutive o rows
        const v4f blo = *(const v4f*)&convb[band * MAXOC + base0];
        const v4f bhi = *(const v4f*)&convb[band * MAXOC + base0 + 4];
        *(v4f*)&Ys[col * YPADN + base0]     = acc[j].lo + blo;
        *(v4f*)&Ys[col * YPADN + base0 + 4] = acc[j].hi + bhi;
    }
    __syncthreads();

    // ---- relu/sigmoid gating + complex multiply + interleaved complex64 out ----
    const int boff  = d_boff[band];
    const int total = 2 /*N_SRC*/ * BW * tlen;
    for (int idx = tid; idx < total; idx += 256) {
        const int t   = idx % tlen;
        const int sf  = idx / tlen;
        const int f   = sf % BW;
        const int src = sf / BW;              // 0..1
        const int rowR = src * BW + f;        // p=0 (real mask) row in relu group
        const int rowI = (2 + src) * BW + f;  // p=1 (imag mask) row in relu group

        const float* yrow = &Ys[t * YPADN];
        float yr = fmaxf(yrow[rowR], 0.f);
        float gr = fmaxf(yrow[rowR + HALFOC], 0.f);
        float yi = fmaxf(yrow[rowI], 0.f);
        float gi = fmaxf(yrow[rowI + HALFOC], 0.f);

        const float mr = yr * (1.f / (1.f + __expf(-gr)));
        const float mi = yi * (1.f / (1.f + __expf(-gi)));

        const size_t cidx = (((size_t)band * BB + b) * MAXBW + f) * TT + (t0 + t);
        const float cr = ctx_r[cidx];
        const float ci = ctx_i[cidx];
        const float re = cr * mr - ci * mi;
        const float im = cr * mi + ci * mr;

        const size_t oidx = (((size_t)b * 2 + src) * ENCDIM + (boff + f)) * TT + (t0 + t);
        out[2 * oidx]     = re;
        out[2 * oidx + 1] = im;
    }
}

// ---------------------------------------------------------------------------
extern "C" void kernel_launch(void* const* d_in, const int* in_sizes, int n_in,
                              void* d_out, int out_size, void* d_ws, size_t ws_size,
                              hipStream_t stream)
{
    const float* sep    = (const float*)d_in[0];
    // d_in[1] = audio_mixture_embedding (unused by the reference)
    const float* ctx_r  = (const float*)d_in[2];
    const float* ctx_i  = (const float*)d_in[3];
    const float* gamma  = (const float*)d_in[4];
    const float* beta   = (const float*)d_in[5];
    const float* convw  = (const float*)d_in[6];
    const float* convb  = (const float*)d_in[7];
    float* out   = (float*)d_out;
    float* stats = (float*)d_ws;            // 248 * 2 floats

    bsrnn_stats_kernel<<<NB * BB, 256, 0, stream>>>(sep, stats);

    const int ntiles = (TT + TILE_T - 1) / TILE_T;   // 16
    const size_t shbytes =
        (size_t)(MAXOC * WPADN + TILE_T * YPADN) * sizeof(float);  // ~100 KB

    // Band groups (contiguous in band index): 0-9 bw=3, 10-21 bw=8, 22-29 bw=16, 30 bw=3
    bsrnn_gemm_mask_kernel<3><<<dim3(ntiles, BB, 10), 256, shbytes, stream>>>(
        sep, ctx_r, ctx_i, gamma, beta, convw, convb, stats, out, /*band_base=*/0);
    bsrnn_gemm_mask_kernel<8><<<dim3(ntiles, BB, 12), 256, shbytes, stream>>>(
        sep, ctx_r, ctx_i, gamma, beta, convw, convb, stats, out, /*band_base=*/10);
    bsrnn_gemm_mask_kernel<16><<<dim3(ntiles, BB, 8), 256, shbytes, stream>>>(
        sep, ctx_r, ctx_i, gamma, beta, convw, convb, stats, out, /*band_base=*/22);
    bsrnn_gemm_mask_kernel<3><<<dim3(ntiles, BB, 1), 256, shbytes, stream>>>(
        sep, ctx_r, ctx_i, gamma, beta, convw, convb, stats, out, /*band_base=*/30);
}